// SPNet_12919261626868
// MI455X (gfx1250) — compile-verified
//
#include <hip/hip_runtime.h>
#include <hip/hip_bf16.h>
#include <math.h>

// ---------------------------------------------------------------------------
// SPNet superpixel clustering on MI455X (gfx1250).
//   GEMM1: dot = f^T @ cf          (3600x256)(256x100)   -> v_wmma_f32_16x16x4_f32
//   GEMM2: U   = fxy @ assocRaw    (258x3600)(3600x100)  -> v_wmma_f32_16x16x4_f32
//   center[:,k] = U[:,k] / colsum[k]   (normalization is linear per column)
// ~30 GFLOP total, all intermediates L2-resident.  f32 WMMA preserves the
// catastrophically-cancelling distance f_sq + c_sq - 2*dot.
//
// Key layout trick: centers and assocRaw are stored ROW-PAIR-INTERLEAVED
//   elem(r,k) -> ((r>>1)*KP + k)*2 + (r&1)
// so a WMMA B fragment (rows ca,ca+1; ca even) is one aligned float2: a single
// ds_load_b64 into an even VGPR pair, zero repacking movs.  The pair layout is
// still byte-contiguous per row-slab, so GLOBAL_LOAD_ASYNC_TO_LDS_B128 stages
// it verbatim (ASYNCcnt path).  Grids cover all 225 M-tiles exactly -> no
// range guards / EXEC toggling in inner loops.
// ---------------------------------------------------------------------------

typedef float v2f __attribute__((ext_vector_type(2)));
typedef float v8f __attribute__((ext_vector_type(8)));

#define B_   8
#define C_   256
#define HH_  60
#define WW_  60
#define N_   3600            // 60*60 = 225 tiles of 16
#define K_   100
#define KP   112             // K padded to 7*16
#define CP   272             // (C+2) padded to 17*16
#define MAX_ITERS 10

// pair-interleaved indices (row r even-pairs share a float2 per column k)
#define CIDX(c, k) ((((c) >> 1) * KP + (k)) * 2 + ((c) & 1))
#define AIDX(n, k) ((((n) >> 1) * KP + (k)) * 2 + ((n) & 1))

// workspace layout (in floats) ----------------------------------------------
#define CENTERS_OFF 0                              // B_*CP*KP      (975 KB)
#define ASSOC_OFF   (CENTERS_OFF + B_*CP*KP)       // B_*N_*KP      (12.9 MB)
#define FSQ_OFF     (ASSOC_OFF   + B_*N_*KP)       // B_*N_
#define CSQ_OFF     (FSQ_OFF     + B_*N_)          // B_*KP
#define COLSUM_OFF  (CSQ_OFF     + B_*KP)          // B_*KP
#define WS_FLOATS   (COLSUM_OFF  + B_*KP)

// --- CDNA5 async global->LDS copy (16B per lane, tracked on ASYNCcnt) ------
__device__ __forceinline__ void async_copy_b128(unsigned lds_off, const void* g) {
  asm volatile("global_load_async_to_lds_b128 %0, %1, off"
               :: "v"(lds_off), "v"((unsigned long long)(uintptr_t)g)
               : "memory");
}
__device__ __forceinline__ void async_wait0() {
  asm volatile("s_wait_asynccnt 0" ::: "memory");
}

// ---------------------------------------------------------------------------
// init: per-pixel squared feature norm
__global__ __launch_bounds__(256) void spnet_init_fsq(
    const float* __restrict__ feat, float* __restrict__ ws)
{
  const int b = blockIdx.y;
  const int n = blockIdx.x * 256 + threadIdx.x;
  if (n >= N_) return;
  const float* featB = feat + (size_t)b * C_ * N_;
  float s = 0.0f;
  #pragma unroll 8
  for (int c = 0; c < C_; ++c) {
    float v = featB[(size_t)c * N_ + n];
    s = fmaf(v, v, s);
  }
  ws[FSQ_OFF + (size_t)b * N_ + n] = s;
}

// init: centers from seeds (pair layout), padded region zeroed
__global__ __launch_bounds__(CP) void spnet_init_centers(
    const float* __restrict__ feat, const int* __restrict__ seeds,
    float* __restrict__ ws)
{
  const int b = blockIdx.x;
  const int c = threadIdx.x;          // 0..271
  const float* featB = feat + (size_t)b * C_ * N_;
  float* centers = ws + CENTERS_OFF + (size_t)b * CP * KP;
  for (int k = 0; k < KP; ++k) {
    float v = 0.0f;
    if (k < K_) {
      const int sh = seeds[(size_t)b * K_ * 2 + k * 2 + 0];
      const int sw = seeds[(size_t)b * K_ * 2 + k * 2 + 1];
      const int sn = sh * WW_ + sw;
      if (c < C_)            v = featB[(size_t)c * N_ + sn];
      else if (c == C_)      v = (float)sh;
      else if (c == C_ + 1)  v = (float)sw;
    }
    centers[CIDX(c, k)] = v;
  }
}

// per-iteration prep: c_sq[k] from current centers, zero colsum
__global__ __launch_bounds__(KP) void spnet_prep(
    const int* __restrict__ n_iter_p, int iter, float* __restrict__ ws)
{
  if (iter >= *n_iter_p) return;
  const int b = blockIdx.x;
  const int k = threadIdx.x;          // 0..111
  const float* centers = ws + CENTERS_OFF + (size_t)b * CP * KP;
  float s = 0.0f;
  #pragma unroll 8
  for (int c = 0; c < C_; ++c) {
    float v = centers[CIDX(c, k)];
    s = fmaf(v, v, s);
  }
  ws[CSQ_OFF    + (size_t)b * KP + k] = s;
  ws[COLSUM_OFF + (size_t)b * KP + k] = 0.0f;
}

// ---------------------------------------------------------------------------
// GEMM1 + elementwise: assocRaw[n,k] and colsum[k].
// 25 blocks x 9 waves cover the 225 M-tiles exactly -> no range guards.
// Centers async-staged into LDS in 64-row chunks (pair layout, contiguous).
__global__ __launch_bounds__(288) void spnet_assoc(
    const float* __restrict__ feat, const int* __restrict__ mask,
    const int* __restrict__ n_iter_p, int iter, float* __restrict__ ws)
{
  if (iter >= *n_iter_p) return;
  const int b    = blockIdx.y;
  const int lane = threadIdx.x & 31;
  const int wave = threadIdx.x >> 5;               // 0..8
  const int l16  = lane & 15;
  const int half = lane >> 4;
  const int n0   = blockIdx.x * 144 + wave * 16;   // wave's M-tile base (<3600)
  const int n    = n0 + l16;                       // A-operand row pixel

  const float* featB   = feat + (size_t)b * C_ * N_;
  const int*   maskB   = mask + (size_t)b * N_;
  const float* centers = ws + CENTERS_OFF + (size_t)b * CP * KP;
  float*       assoc   = ws + ASSOC_OFF   + (size_t)b * N_ * KP;
  const float* fsq     = ws + FSQ_OFF     + (size_t)b * N_;
  const float* csq     = ws + CSQ_OFF     + (size_t)b * KP;
  float*       colsum  = ws + COLSUM_OFF  + (size_t)b * KP;

  __shared__ __align__(16) float ldsC[64 * KP];    // centers chunk: 28.7 KB
  __shared__ float ldsColsum[KP];

  if (threadIdx.x < KP) ldsColsum[threadIdx.x] = 0.0f;
  const unsigned ldsC_base = (unsigned)(uintptr_t)&ldsC[0];

  v8f acc[7];
  #pragma unroll
  for (int kt = 0; kt < 7; ++kt) {
    v8f z = {0.f, 0.f, 0.f, 0.f, 0.f, 0.f, 0.f, 0.f};
    acc[kt] = z;
  }

  for (int c0 = 0; c0 < C_; c0 += 64) {
    __syncthreads();
    // async-stage centers rows c0..c0+63 (contiguous 28672B slab, pair layout)
    {
      const char* g = (const char*)(centers + (size_t)c0 * KP);
      for (int t = threadIdx.x; t < (64 * KP * 4) / 16; t += 288)
        async_copy_b128(ldsC_base + t * 16, g + t * 16);
      async_wait0();
    }
    __syncthreads();

    if (c0 + 64 < C_)   // speculative prefetch of next A chunk
      __builtin_prefetch(&featB[(size_t)(c0 + 64 + 2 * half) * N_ + n], 0, 0);

    #pragma unroll 4
    for (int c = 0; c < 64; c += 4) {
      const int ca = c + 2 * half;                 // VGPR0 column (even)
      const int cp = (ca >> 1) * KP;               // pair-row base in LDS
      v2f av;
      av[0] = featB[(size_t)(c0 + ca)     * N_ + n];
      av[1] = featB[(size_t)(c0 + ca + 1) * N_ + n];
      #pragma unroll
      for (int kt = 0; kt < 7; ++kt) {
        // rows (ca,ca+1) at column k are one aligned float2: ds_load_b64
        const v2f bv = *(const v2f*)&ldsC[(cp + kt * 16 + l16) * 2];
        acc[kt] = __builtin_amdgcn_wmma_f32_16x16x4_f32(
            false, av, false, bv, (short)0, acc[kt], false, false);
      }
    }
  }

  // epilogue: distance -> exp(-sqrt) -> masked assoc + column sums
  float fsq_v[8], hh_v[8], wv_v[8], mk_v[8];
  #pragma unroll
  for (int v = 0; v < 8; ++v) {
    const int nn = n0 + v + 8 * half;              // C/D row: M = v + 8*half
    fsq_v[v] = fsq[nn];
    mk_v[v]  = (maskB[nn] == 1) ? 1.0f : 0.0f;
    hh_v[v]  = (float)(nn / WW_);
    wv_v[v]  = (float)(nn % WW_);
  }

  #pragma unroll
  for (int kt = 0; kt < 7; ++kt) {
    const int k = kt * 16 + l16;
    const float kmask = (k < K_) ? 1.0f : 0.0f;
    const float csqk = csq[k];
    const v2f  cxy = *(const v2f*)&centers[((C_ >> 1) * KP + k) * 2]; // rows 256,257
    float lsum = 0.0f;
    #pragma unroll
    for (int v = 0; v < 8; v += 2) {               // row pair (nn, nn+1)
      v2f a2;
      #pragma unroll
      for (int u = 0; u < 2; ++u) {
        const float fd = fsq_v[v + u] + csqk - 2.0f * acc[kt][v + u];
        const float dh = hh_v[v + u] - cxy[0];
        const float dw = wv_v[v + u] - cxy[1];
        const float tot = sqrtf(fmaxf(fd + (dh * dh + dw * dw) * 0.01f, 0.0f) + 1e-12f);
        a2[u] = __expf(-tot) * mk_v[v + u] * kmask;
      }
      const int nn = n0 + v + 8 * half;            // even -> aligned pair store
      *(v2f*)&assoc[AIDX(nn, k)] = a2;             // global_store_b64
      lsum += a2[0] + a2[1];
    }
    atomicAdd(&ldsColsum[k], lsum);                // ds_add_f32
  }
  __syncthreads();
  if (threadIdx.x < KP) {
    const float s = ldsColsum[threadIdx.x];
    if (s != 0.0f) atomicAdd(&colsum[threadIdx.x], s); // global_atomic_add_f32
  }
}

// ---------------------------------------------------------------------------
// GEMM2: U = fxy_padded @ assocRaw, then center = U / colsum.
// ct = blockIdx.x picks a 16-row c tile, waves 0..6 each own a 16-col k tile.
// fxy tile staged transposed (pad 62 -> b64 A reads, conflict-free); assoc
// chunk (contiguous 26880B slab, pair layout) async-staged into LDS.
__global__ __launch_bounds__(256) void spnet_update(
    const float* __restrict__ feat, const int* __restrict__ n_iter_p,
    int iter, float* __restrict__ ws)
{
  if (iter >= *n_iter_p) return;
  const int ct   = blockIdx.x;        // 0..16
  const int b    = blockIdx.y;
  const int lane = threadIdx.x & 31;
  const int wave = threadIdx.x >> 5;  // kt = wave for waves 0..6
  const int l16  = lane & 15;
  const int half = lane >> 4;

  const float* featB  = feat + (size_t)b * C_ * N_;
  const float* assoc  = ws + ASSOC_OFF   + (size_t)b * N_ * KP;
  const float* colsum = ws + COLSUM_OFF  + (size_t)b * KP;
  float*       centers= ws + CENTERS_OFF + (size_t)b * CP * KP;

  __shared__ __align__(16) float ldsA[16 * 62];    // fxy tile, padded rows
  __shared__ __align__(16) float ldsS[60 * KP];    // assoc chunk (26.9 KB)
  const unsigned ldsS_base = (unsigned)(uintptr_t)&ldsS[0];

  v8f acc = {0.f, 0.f, 0.f, 0.f, 0.f, 0.f, 0.f, 0.f};

  for (int nchunk = 0; nchunk < N_ / 60; ++nchunk) {
    const int n0 = nchunk * 60;
    __syncthreads();
    // fxy rows ct*16..ct*16+15, cols n0..n0+59 (coords / zero-pad rows)
    for (int idx = threadIdx.x; idx < 16 * 60; idx += 256) {
      const int r = idx / 60, j = idx % 60;
      const int c = ct * 16 + r;
      const int nn = n0 + j;
      float v;
      if (c < C_)           v = featB[(size_t)c * N_ + nn];
      else if (c == C_)     v = (float)(nn / WW_);
      else if (c == C_ + 1) v = (float)(nn % WW_);
      else                  v = 0.0f;
      ldsA[r * 62 + j] = v;
    }
    // assoc rows n0..n0+59 (pair layout, contiguous slab) -> async copy
    {
      const char* g = (const char*)(assoc + (size_t)n0 * KP);
      for (int t = threadIdx.x; t < (60 * KP * 4) / 16; t += 256)
        async_copy_b128(ldsS_base + t * 16, g + t * 16);
      async_wait0();
    }
    __syncthreads();

    if (wave < 7) {
      #pragma unroll 5
      for (int nn = 0; nn < 60; nn += 4) {
        const int na = nn + 2 * half;              // VGPR0 n column (even)
        const v2f av = *(const v2f*)&ldsA[l16 * 62 + na];          // ds_load_b64
        const v2f bv = *(const v2f*)&ldsS[((na >> 1) * KP + wave * 16 + l16) * 2];
        acc = __builtin_amdgcn_wmma_f32_16x16x4_f32(
            false, av, false, bv, (short)0, acc, false, false);
      }
    }
  }

  if (wave < 7) {
    const int k = wave * 16 + l16;
    const float s = colsum[k];
    const bool kok = (k < K_) && (s != 0.0f);
    const float inv = kok ? (1.0f / s) : 0.0f;
    #pragma unroll
    for (int v = 0; v < 8; v += 2) {               // row pair (c, c+1)
      const int c = ct * 16 + v + 8 * half;        // even
      v2f o;
      o[0] = kok ? acc[v]     * inv : 0.0f;
      o[1] = kok ? acc[v + 1] * inv : 0.0f;
      *(v2f*)&centers[CIDX(c, k)] = o;             // global_store_b64
    }
  }
}

// ---------------------------------------------------------------------------
__global__ __launch_bounds__(256) void spnet_emit(
    const float* __restrict__ ws, float* __restrict__ out)
{
  const int b = blockIdx.y;
  const int idx = blockIdx.x * 256 + threadIdx.x;   // 0..C_*K_-1 exactly
  const int c = idx / K_;
  const int k = idx % K_;
  out[(size_t)b * C_ * K_ + idx] =
      ws[CENTERS_OFF + (size_t)b * CP * KP + CIDX(c, k)];
}

// ---------------------------------------------------------------------------
extern "C" void kernel_launch(void* const* d_in, const int* in_sizes, int n_in,
                              void* d_out, int out_size, void* d_ws, size_t ws_size,
                              hipStream_t stream) {
  const float* feat   = (const float*)d_in[0];   // (8,256,60,60)
  const int*   mask   = (const int*)  d_in[1];   // (8,60,60)
  const int*   seeds  = (const int*)  d_in[2];   // (8,100,2)
  const int*   niter  = (const int*)  d_in[3];   // scalar (=10 in setup)
  float*       out    = (float*)d_out;           // (8,256,100)
  float*       ws     = (float*)d_ws;            // needs WS_FLOATS*4 ~ 14 MB

  spnet_init_fsq    <<<dim3((N_ + 255) / 256, B_), 256, 0, stream>>>(feat, ws);
  spnet_init_centers<<<dim3(B_),                   CP,  0, stream>>>(feat, seeds, ws);

  for (int it = 0; it < MAX_ITERS; ++it) {
    spnet_prep  <<<dim3(B_),              KP,  0, stream>>>(niter, it, ws);
    spnet_assoc <<<dim3(N_ / 144, B_),    288, 0, stream>>>(feat, mask, niter, it, ws);
    spnet_update<<<dim3(CP / 16, B_),     256, 0, stream>>>(feat, niter, it, ws);
  }

  spnet_emit<<<dim3((C_ * K_) / 256, B_), 256, 0, stream>>>(ws, out);
}